// AlphaComposition_16956530884651
// MI455X (gfx1250) — compile-verified
//
#include <hip/hip_runtime.h>

// Alpha composition over D=32 layers: purely HBM-bandwidth-bound streaming scan.
// Dims fixed per reference: B=4, D=32, C=3, H=256, W=384.
#define B_  4
#define D_  32
#define HW_ (256 * 384)   // 98304, divisible by 4
#define VEC 4

typedef float f4 __attribute__((ext_vector_type(VEC)));

static __device__ __forceinline__ f4 vclamp_min(f4 x, float c) {
    f4 r;
    r.x = fmaxf(x.x, c);
    r.y = fmaxf(x.y, c);
    r.z = fmaxf(x.z, c);
    r.w = fmaxf(x.w, c);
    return r;
}

__global__ __launch_bounds__(256) void alpha_composition_kernel(
    const float* __restrict__ src,    // [B, D, 3, H, W]
    const float* __restrict__ alpha,  // [B, D, 1, H, W]
    float* __restrict__ out)          // [B, 3, H, W]
{
    const int vpb = HW_ / VEC;                 // float4 groups per (b,d,c) plane
    const int t   = blockIdx.x * blockDim.x + threadIdx.x;
    if (t >= B_ * vpb) return;

    const int b = t / vpb;
    const int v = t - b * vpb;
    const int p = v * VEC;                     // pixel index within plane

    // Base pointers for this (b, pixel-group); strides in f4 units.
    const f4* __restrict__ aPtr = (const f4*)(alpha + (size_t)b * D_ * HW_ + p);
    const f4* __restrict__ sPtr = (const f4*)(src   + (size_t)b * D_ * 3 * HW_ + p);
    const int aStrideD = HW_ / VEC;            // alpha: layer-to-layer
    const int sStrideC = HW_ / VEC;            // src:   channel-to-channel
    const int sStrideD = 3 * HW_ / VEC;        // src:   layer-to-layer

    f4 T    = {1.0f, 1.0f, 1.0f, 1.0f};        // transmittance per pixel
    f4 acc0 = {0.0f, 0.0f, 0.0f, 0.0f};
    f4 acc1 = {0.0f, 0.0f, 0.0f, 0.0f};
    f4 acc2 = {0.0f, 0.0f, 0.0f, 0.0f};

    // ---- 2-stage software pipeline: prefetch layer d+1 while computing d ----
    // All traffic is touch-once -> non-temporal (TH=NT) b128 loads.
    f4 aC  = __builtin_nontemporal_load(aPtr);
    f4 s0C = __builtin_nontemporal_load(sPtr);
    f4 s1C = __builtin_nontemporal_load(sPtr + sStrideC);
    f4 s2C = __builtin_nontemporal_load(sPtr + 2 * sStrideC);

#pragma unroll
    for (int d = 0; d < D_; ++d) {
        f4 aN, s0N, s1N, s2N;
        if (d + 1 < D_) {
            const f4* ap = aPtr + (d + 1) * aStrideD;
            const f4* sp = sPtr + (d + 1) * sStrideD;
            aN  = __builtin_nontemporal_load(ap);
            s0N = __builtin_nontemporal_load(sp);
            s1N = __builtin_nontemporal_load(sp + sStrideC);
            s2N = __builtin_nontemporal_load(sp + 2 * sStrideC);
        }

        const f4 a = vclamp_min(aC, 1e-4f);
        const f4 w = T * a;                    // weight for this layer
        acc0 += w * s0C;                       // contracts to v_fma_f32
        acc1 += w * s1C;
        acc2 += w * s2C;
        T *= (1.0f - a);                       // update transmittance

        if (d + 1 < D_) {
            aC = aN; s0C = s0N; s1C = s1N; s2C = s2N;
        }
    }

    f4* __restrict__ oPtr = (f4*)(out + (size_t)b * 3 * HW_ + p);
    __builtin_nontemporal_store(acc0, oPtr);
    __builtin_nontemporal_store(acc1, oPtr + sStrideC);
    __builtin_nontemporal_store(acc2, oPtr + 2 * sStrideC);
}

extern "C" void kernel_launch(void* const* d_in, const int* in_sizes, int n_in,
                              void* d_out, int out_size, void* d_ws, size_t ws_size,
                              hipStream_t stream) {
    const float* src   = (const float*)d_in[0];  // [4,32,3,256,384] fp32
    const float* alpha = (const float*)d_in[1];  // [4,32,1,256,384] fp32
    float*       out   = (float*)d_out;          // [4,3,256,384]    fp32

    const int totalThreads = B_ * (HW_ / VEC);   // 98304 threads (wave32 x 3072)
    const int block = 256;                       // 8 waves per workgroup
    const int grid  = (totalThreads + block - 1) / block;

    alpha_composition_kernel<<<grid, block, 0, stream>>>(src, alpha, out);
}